// BasicDeconvolutionBlockup_25898652795314
// MI455X (gfx1250) — compile-verified
//
#include <hip/hip_runtime.h>

typedef __attribute__((ext_vector_type(2))) float v2f;
typedef __attribute__((ext_vector_type(8))) float v8f;

#define N_IN    250000
#define N_OUTV  500000
#define INC     256
#define OUTC    128
#define KVOL    8
#define ROWS_PER_BLOCK 128
#define LDS_STRIDE     260   // 256 + 4 pad: 16B-aligned rows, conflict-free frag reads

// ---------------------------------------------------------------------------
// Zero-init d_out accumulator and the BN stats scratch (harness poisons both).
// ---------------------------------------------------------------------------
__global__ void zero_kernel(float* __restrict__ out, double* __restrict__ ws) {
    size_t i      = (size_t)blockIdx.x * blockDim.x + threadIdx.x;
    size_t stride = (size_t)gridDim.x * blockDim.x;
    size_t n4     = (size_t)N_OUTV * OUTC / 4;
    float4* o4 = (float4*)out;
    float4 z; z.x = 0.f; z.y = 0.f; z.z = 0.f; z.w = 0.f;
    for (size_t j = i; j < n4; j += stride) o4[j] = z;
    if (i < 2 * OUTC) ws[i] = 0.0;
}

// ---------------------------------------------------------------------------
// Sparse transposed-conv: per block, stage 128 rows of x in LDS once (via the
// CDNA5 async global->LDS path), then for each of the 8 kernel offsets run a
// 128x128 GEMM tile with V_WMMA_F32_16X16X4_F32 and scatter-add rows into out
// with native f32 atomics (half-wave-coalesced to 64B segments).
// ---------------------------------------------------------------------------
__global__ void __launch_bounds__(256)
spconv_wmma_kernel(const float* __restrict__ x, const float* __restrict__ W,
                   const int* __restrict__ out_idx, float* __restrict__ out) {
    extern __shared__ float lds[];  // ROWS_PER_BLOCK * LDS_STRIDE floats
    const int tid     = threadIdx.x;
    const int baseRow = blockIdx.x * ROWS_PER_BLOCK;

    // Cooperative tile fill: async global->LDS (GLOBAL_LOAD_ASYNC_TO_LDS_B128,
    // ASYNCcnt-tracked, no VGPR round-trip). OOB pad rows get zeros via DS.
    for (int t = tid; t < ROWS_PER_BLOCK * (INC / 4); t += 256) {
        int r  = t >> 6;      // row within tile (64 float4 per row)
        int c4 = t & 63;      // float4 column
        int g  = baseRow + r;
        unsigned ldsOff = (unsigned)(((unsigned)r * LDS_STRIDE + (unsigned)c4 * 4) * sizeof(float));
        if (g < N_IN) {
            unsigned gOff = (unsigned)g * (INC * (unsigned)sizeof(float)) + (unsigned)c4 * 16u;
            asm volatile("global_load_async_to_lds_b128 %0, %1, %2"
                         :: "v"(ldsOff), "v"(gOff), "s"(x)
                         : "memory");
        } else {
            float4 zz; zz.x = 0.f; zz.y = 0.f; zz.z = 0.f; zz.w = 0.f;
            *(float4*)(&lds[r * LDS_STRIDE + c4 * 4]) = zz;
        }
    }
#if __has_builtin(__builtin_amdgcn_s_wait_asynccnt)
    __builtin_amdgcn_s_wait_asynccnt(0);
#else
    asm volatile("s_wait_asynccnt 0x0" ::: "memory");
#endif
    __syncthreads();

    const int lane = tid & 31;      // wave32
    const int wave = tid >> 5;      // 8 waves per block
    const int n0   = wave * 16;     // output column tile
    const int nl   = lane & 15;     // column within tile / A row (M)
    const int hi   = lane >> 4;     // lane half
    const int koff = hi * 2;        // A/B VGPR K layout: VGPR0 = K0|K2, VGPR1 = K1|K3

    const v8f zero8 = {0.f, 0.f, 0.f, 0.f, 0.f, 0.f, 0.f, 0.f};

    for (int k = 0; k < KVOL; ++k) {
        v8f acc[8];
#pragma unroll
        for (int mt = 0; mt < 8; ++mt) acc[mt] = zero8;

        // B operand base: W[k][:, n0+nl], row stride OUTC floats (L2-resident).
        const float* Wk = W + (size_t)k * INC * OUTC + n0 + nl;

        for (int kk = 0; kk < INC / 4; ++kk) {
            const int krow = kk * 4 + koff;
            v2f b;
            b.x = Wk[(size_t)krow * OUTC];
            b.y = Wk[(size_t)(krow + 1) * OUTC];
#pragma unroll
            for (int mt = 0; mt < 8; ++mt) {
                // A fragment: x[tileRow=mt*16+nl][kk*4+koff .. +1], 8B-aligned LDS read
                v2f a = *(const v2f*)(&lds[(mt * 16 + nl) * LDS_STRIDE + kk * 4 + koff]);
                acc[mt] = __builtin_amdgcn_wmma_f32_16x16x4_f32(
                    /*neg_a=*/false, a, /*neg_b=*/false, b,
                    /*c_mod=*/(short)0, acc[mt],
                    /*reuse_a=*/false, /*reuse_b=*/false);
            }
        }

        // Scatter-add: C/D layout => VGPR r holds M=r (lanes 0-15) / M=r+8 (16-31).
        const int* idxk = out_idx + (size_t)k * N_IN;
#pragma unroll 1
        for (int mt = 0; mt < 8; ++mt) {
#pragma unroll
            for (int r = 0; r < 8; ++r) {
                int g = baseRow + mt * 16 + r + hi * 8;
                if (g < N_IN) {
                    int di = idxk[g];
                    unsafeAtomicAdd(out + (size_t)di * OUTC + n0 + nl, acc[mt][r]);
                }
            }
        }
    }
}

// ---------------------------------------------------------------------------
// BN batch stats: per-column sum / sum-of-squares (double) -> ws[0..255].
// ---------------------------------------------------------------------------
__global__ void stats_kernel(const float* __restrict__ out, double* __restrict__ ws) {
    const int col = threadIdx.x;  // 128 threads = 128 columns, coalesced rows
    double s1 = 0.0, s2 = 0.0;
    for (int row = blockIdx.x; row < N_OUTV; row += gridDim.x) {
        double v = (double)out[(size_t)row * OUTC + col];
        s1 += v;
        s2 += v * v;
    }
    unsafeAtomicAdd(&ws[col], s1);
    unsafeAtomicAdd(&ws[OUTC + col], s2);
}

// ---------------------------------------------------------------------------
// BN normalize in place: y = (out - mean) * rsqrt(var + eps) * gamma + beta.
// ---------------------------------------------------------------------------
__global__ void bn_kernel(float* __restrict__ out, const double* __restrict__ ws,
                          const float* __restrict__ gamma, const float* __restrict__ beta) {
    size_t i      = (size_t)blockIdx.x * blockDim.x + threadIdx.x;
    size_t stride = (size_t)gridDim.x * blockDim.x;
    const size_t n    = (size_t)N_OUTV * OUTC;
    const double invN = 1.0 / (double)N_OUTV;
    for (; i < n; i += stride) {
        int col     = (int)(i & (OUTC - 1));
        double mean = ws[col] * invN;
        double var  = ws[OUTC + col] * invN - mean * mean;
        float scale = gamma[col] * rsqrtf((float)var + 1e-5f);
        out[i] = (float)((double)out[i] - mean) * scale + beta[col];
    }
}

extern "C" void kernel_launch(void* const* d_in, const int* in_sizes, int n_in,
                              void* d_out, int out_size, void* d_ws, size_t ws_size,
                              hipStream_t stream) {
    const float* x      = (const float*)d_in[0];
    const float* W      = (const float*)d_in[1];
    const float* gamma  = (const float*)d_in[2];
    const float* beta   = (const float*)d_in[3];
    const int*   outidx = (const int*)d_in[4];
    float*  out = (float*)d_out;
    double* ws  = (double*)d_ws;

    zero_kernel<<<4096, 256, 0, stream>>>(out, ws);

    const int nBlocks = (N_IN + ROWS_PER_BLOCK - 1) / ROWS_PER_BLOCK;
    const size_t smem = (size_t)ROWS_PER_BLOCK * LDS_STRIDE * sizeof(float); // 130 KB < 320 KB
    spconv_wmma_kernel<<<nBlocks, 256, smem, stream>>>(x, W, outidx, out);

    stats_kernel<<<2048, 128, 0, stream>>>(out, ws);
    bn_kernel<<<8192, 256, 0, stream>>>(out, ws, gamma, beta);
}